// TransportModule_81398220194374
// MI455X (gfx1250) — compile-verified
//
#include <hip/hip_runtime.h>
#include <hip/hip_bf16.h>
#include <stdint.h>

typedef float v2f __attribute__((ext_vector_type(2)));
typedef float v8f __attribute__((ext_vector_type(8)));

#define B_  16
#define N_  8192
#define D_  128
#define P_  128
#define BN_ (B_ * N_)            // 131072
#define SORT_THREADS 256

// ---------------------------------------------------------------- RNG helpers
__device__ __forceinline__ uint32_t pcg_hash(uint32_t v) {
    uint32_t s = v * 747796405u + 2891336453u;
    uint32_t w = ((s >> ((s >> 28) + 4u)) ^ s) * 277803737u;
    return (w >> 22) ^ w;
}

// ---------------------------------------------------------------- K1: thetas
// One block per projection row p (128 threads). Box-Muller gaussian from a
// hash, then row-normalize: theta = g / max(||g||, 1e-8).
__global__ void gen_theta_kernel(float* __restrict__ theta) {
    const int p = blockIdx.x;
    const int t = threadIdx.x;                 // 0..127
    const uint32_t i = (uint32_t)(p * D_ + t);

    uint32_t ra = pcg_hash(2u * i + 42u * 0x9E3779B9u);
    uint32_t rb = pcg_hash(2u * i + 1u + 42u * 0x9E3779B9u);
    float u1 = ((float)ra + 1.0f) * 2.3283064365386963e-10f;   // (0,1]
    float u2 = (float)rb * 2.3283064365386963e-10f;            // [0,1)
    float g = sqrtf(-2.0f * __logf(u1)) * __cosf(6.28318530718f * u2);

    __shared__ float red[D_];
    red[t] = g * g;
    __syncthreads();
    for (int s = D_ / 2; s > 0; s >>= 1) {
        if (t < s) red[t] += red[t + s];
        __syncthreads();
    }
    float nrm = fmaxf(sqrtf(red[0]), 1e-8f);
    theta[p * D_ + t] = g / nrm;
}

// ---------------------------------------------------------------- K2: project
// C[bn, p] = sum_k src[bn, k] * theta[p, k], stored transposed into
// proj[p * BN + bn]. One 16x16 output tile per wave, K-loop of 32
// v_wmma_f32_16x16x4_f32 ops (full fp32, matches reference precision).
__global__ void project_kernel(const float* __restrict__ x,
                               const float* __restrict__ y,
                               const float* __restrict__ theta,
                               float* __restrict__ xproj,
                               float* __restrict__ yproj) {
    const int wave = threadIdx.x >> 5;
    const int lane = threadIdx.x & 31;
    const long tile = (long)blockIdx.x * 8 + wave;   // 0 .. 2*65536-1
    const int mat = (int)(tile >> 16);               // 0 = x, 1 = y
    const int t = (int)(tile & 65535);
    const int tp = t & 7;                            // P tile (8 of them)
    const long tbn = t >> 3;                         // BN tile (8192 of them)
    const int p0 = tp * 16;
    const long bn0 = tbn * 16;

    const float* __restrict__ src = mat ? y : x;
    float* __restrict__ dst = mat ? yproj : xproj;

    const int m = lane & 15;       // A: row m (bn);  B: col n (p)
    const int kh = lane >> 4;      // which K-pair this half-wave holds

    const float* arow = src + (bn0 + m) * D_;
    const float* brow = theta + (p0 + m) * D_;       // B[k][n] = theta[n][k]

    v8f c = {};
    #pragma unroll
    for (int kk = 0; kk < 32; ++kk) {
        const int k0 = kk * 4 + kh * 2;
        v2f a; a.x = arow[k0]; a.y = arow[k0 + 1];
        v2f b; b.x = brow[k0]; b.y = brow[k0 + 1];
        c = __builtin_amdgcn_wmma_f32_16x16x4_f32(false, a, false, b,
                                                  (short)0, c, false, false);
    }

    // C(mrow, n) -> proj[(p0+n)*BN + bn0 + mrow]   (transposed store)
    const int n = m;
    const int mbase = kh * 8;
    #pragma unroll
    for (int r = 0; r < 8; ++r) {
        dst[(long)(p0 + n) * BN_ + bn0 + mbase + r] = c[r];
    }
}

// ---------------------------------------------------------------- K3a: sort y
// One (p,b) row of 8192 floats per block, ascending bitonic sort in LDS.
__global__ void sort_y_kernel(float* __restrict__ yproj) {
    __shared__ float key[N_];                        // 32 KB
    const long base = (long)blockIdx.x * N_;
    const int tid = threadIdx.x;

    for (int i = tid; i < N_; i += SORT_THREADS) key[i] = yproj[base + i];
    __syncthreads();

    for (int k = 2; k <= N_; k <<= 1) {
        for (int j = k >> 1; j > 0; j >>= 1) {
            for (int i = tid; i < N_; i += SORT_THREADS) {
                const int ixj = i ^ j;
                if (ixj > i) {
                    const bool up = ((i & k) == 0);
                    const float a = key[i], b = key[ixj];
                    if ((a > b) == up) { key[i] = b; key[ixj] = a; }
                }
            }
            __syncthreads();
        }
    }

    for (int i = tid; i < N_; i += SORT_THREADS) yproj[base + i] = key[i];
}

// ------------------------------------------------------ K3b: sort x + scatter
// Sort (x value, original index) pairs per row, then scatter
// diff[x_idx[k]] = y_sorted[k] - x_sorted[k] in-place over the xproj row.
__global__ void sort_x_diff_kernel(float* __restrict__ xproj,
                                   const float* __restrict__ ysorted) {
    __shared__ float key[N_];                        // 32 KB
    __shared__ int   idx[N_];                        // 32 KB
    const long base = (long)blockIdx.x * N_;
    const int tid = threadIdx.x;

    for (int i = tid; i < N_; i += SORT_THREADS) {
        key[i] = xproj[base + i];
        idx[i] = i;
    }
    __syncthreads();

    for (int k = 2; k <= N_; k <<= 1) {
        for (int j = k >> 1; j > 0; j >>= 1) {
            for (int i = tid; i < N_; i += SORT_THREADS) {
                const int ixj = i ^ j;
                if (ixj > i) {
                    const bool up = ((i & k) == 0);
                    const float a = key[i], b = key[ixj];
                    if ((a > b) == up) {
                        key[i] = b; key[ixj] = a;
                        const int ti = idx[i]; idx[i] = idx[ixj]; idx[ixj] = ti;
                    }
                }
            }
            __syncthreads();
        }
    }

    for (int i = tid; i < N_; i += SORT_THREADS) {
        const float d = ysorted[base + i] - key[i];
        xproj[base + idx[i]] = d;                    // global scatter
    }
}

// ---------------------------------------------------------------- K4: recombine
// out[bn, dd] = x[bn, dd] + (1/P) * sum_p diff[p*BN + bn] * theta[p, dd]
__global__ void recombine_kernel(const float* __restrict__ x,
                                 const float* __restrict__ theta,
                                 const float* __restrict__ diff,
                                 float* __restrict__ out) {
    const int wave = threadIdx.x >> 5;
    const int lane = threadIdx.x & 31;
    const long tile = (long)blockIdx.x * 8 + wave;   // 0 .. 65535
    const int tdd = (int)(tile & 7);
    const long tbn = tile >> 3;
    const int dd0 = tdd * 16;
    const long bn0 = tbn * 16;

    const int m = lane & 15;       // A row (bn) / B col (dd)
    const int kh = lane >> 4;
    const int n = m;

    v8f c = {};
    #pragma unroll
    for (int kk = 0; kk < 32; ++kk) {
        const int k0 = kk * 4 + kh * 2;
        v2f a; a.x = diff[(long)k0 * BN_ + bn0 + m];
               a.y = diff[(long)(k0 + 1) * BN_ + bn0 + m];
        v2f b; b.x = theta[k0 * D_ + dd0 + n];
               b.y = theta[(k0 + 1) * D_ + dd0 + n];
        c = __builtin_amdgcn_wmma_f32_16x16x4_f32(false, a, false, b,
                                                  (short)0, c, false, false);
    }

    const float invP = 1.0f / (float)P_;
    #pragma unroll
    for (int r = 0; r < 8; ++r) {
        const long row = bn0 + kh * 8 + r;
        const long off = row * D_ + dd0 + n;
        out[off] = x[off] + c[r] * invP;
    }
}

// ---------------------------------------------------------------- launch
extern "C" void kernel_launch(void* const* d_in, const int* in_sizes, int n_in,
                              void* d_out, int out_size, void* d_ws, size_t ws_size,
                              hipStream_t stream) {
    (void)in_sizes; (void)n_in; (void)out_size; (void)ws_size;

    const float* x = (const float*)d_in[0];     // (16, 8192, 128) f32
    const float* y = (const float*)d_in[1];     // (16, 8192, 128) f32
    float* out = (float*)d_out;                 // (16, 8192, 128) f32

    float* theta = (float*)d_ws;                // P*D = 16384 floats (64 KB)
    float* xdiff = theta + (P_ * D_);           // P*BN floats (64 MB): xproj -> diff
    float* yproj = out;                         // y projections live in d_out
                                                // (dead before recombine writes it)

    // K1: projection directions
    gen_theta_kernel<<<P_, D_, 0, stream>>>(theta);

    // K2: x and y projections (2 * 65536 wave-tiles, 8 waves per block)
    project_kernel<<<(2 * (BN_ / 16) * (P_ / 16)) / 8, 256, 0, stream>>>(
        x, y, theta, xdiff, yproj);

    // K3: per-(p,b) sorts — 2048 rows of 8192
    sort_y_kernel<<<P_ * B_, SORT_THREADS, 0, stream>>>(yproj);
    sort_x_diff_kernel<<<P_ * B_, SORT_THREADS, 0, stream>>>(xdiff, yproj);

    // K4: recombine + residual add (65536 wave-tiles, 8 waves per block)
    recombine_kernel<<<((BN_ / 16) * (D_ / 16)) / 8, 256, 0, stream>>>(
        x, theta, xdiff, out);
}